// Attention_50268297232537
// MI455X (gfx1250) — compile-verified
//
#include <hip/hip_runtime.h>

typedef __attribute__((ext_vector_type(2))) float v2f;
typedef __attribute__((ext_vector_type(8))) float v8f;

static constexpr int B = 64, L = 1024, H2 = 1024;
static constexpr int LCHUNKS = 16;          // phase-4 L split for occupancy
static constexpr int LSUB = L / LCHUNKS;    // 64

// ---------------------------------------------------------------------------
// Phase 1: dec[i][j] = sum_k s[i][k] * W[j][k] + bias[j]
// One wave per 16x16 tile, V_WMMA_F32_16X16X4_F32, K-loop of 256 steps.
// A-frag (16x4 f32): lane<16 -> M=lane, K={0,1}; lane>=16 -> M=lane-16, K={2,3}
// B-frag (4x16 f32): lane<16 -> N=lane, K={0,1}; lane>=16 -> N=lane-16, K={2,3}
// C/D (16x16 f32):  lanes0-15: M=r, N=lane; lanes16-31: M=8+r, N=lane-16
// ---------------------------------------------------------------------------
__global__ __launch_bounds__(32)
void dec_gemm_wmma(const float* __restrict__ s, const float* __restrict__ W,
                   const float* __restrict__ bias, float* __restrict__ dec) {
    const int col0 = blockIdx.x * 16;   // N tile (0..63)
    const int row0 = blockIdx.y * 16;   // M tile (0..3)
    const int lane = threadIdx.x;
    const int half = lane >> 4;         // 0: K base 0, 1: K base 2
    const int l15  = lane & 15;

    // bias[j] is constant along M -> every C register of this lane gets b[col]
    const float bj = bias[col0 + l15];
    v8f c;
#pragma unroll
    for (int r = 0; r < 8; ++r) c[r] = bj;

    const float* arow = s + (row0 + l15) * H2 + half * 2;  // A: row M=l15
    const float* brow = W + (col0 + l15) * H2 + half * 2;  // B: col N=l15 (W row-major)

#pragma unroll 4
    for (int k = 0; k < H2; k += 4) {
        v2f a  = *(const v2f*)(arow + k);   // 8B-aligned (k,half*2 even)
        v2f bb = *(const v2f*)(brow + k);
        c = __builtin_amdgcn_wmma_f32_16x16x4_f32(
                /*neg_a=*/false, a, /*neg_b=*/false, bb,
                /*c_mod=*/(short)0, c, /*reuse_a=*/false, /*reuse_b=*/false);
    }

#pragma unroll
    for (int r = 0; r < 8; ++r) {
        const int m = half * 8 + r;
        dec[(row0 + m) * H2 + col0 + l15] = c[r];
    }
}

// ---------------------------------------------------------------------------
// Phase 2: e[b][l] = sum_h tanh(enc_feat[b*L+l][h] + dec[b][h]) * v[h]
// One wave per row; 8 float4 loads per lane (coalesced b128), wave32 reduce.
// Streams 256 MB -> bandwidth-bound.
// ---------------------------------------------------------------------------
__global__ __launch_bounds__(256)
void score_kernel(const float* __restrict__ encf, const float* __restrict__ dec,
                  const float* __restrict__ v, float* __restrict__ e) {
    const int warp = threadIdx.x >> 5;
    const int lane = threadIdx.x & 31;
    const long long row = (long long)blockIdx.x * 8 + warp;  // 0..65535
    const int b = (int)(row >> 10);

    const float4* er = (const float4*)(encf + row * H2);
    const float4* dr = (const float4*)(dec + (long long)b * H2);
    const float4* vr = (const float4*)v;

    float sum = 0.f;
#pragma unroll
    for (int i = 0; i < 8; ++i) {
        const int idx = i * 32 + lane;
        float4 x = er[idx], d = dr[idx], w = vr[idx];
        sum += tanhf(x.x + d.x) * w.x + tanhf(x.y + d.y) * w.y +
               tanhf(x.z + d.z) * w.z + tanhf(x.w + d.w) * w.w;
    }
#pragma unroll
    for (int o = 16; o > 0; o >>= 1) sum += __shfl_down(sum, o, 32);
    if (lane == 0) e[row] = sum;
}

// ---------------------------------------------------------------------------
// Phase 3: masked softmax over L per batch (Z of the unmasked softmax cancels)
// ---------------------------------------------------------------------------
__global__ __launch_bounds__(256)
void softmax_kernel(const float* __restrict__ e, const float* __restrict__ mask,
                    float* __restrict__ attn) {
    const int b   = blockIdx.x;
    const int tid = threadIdx.x;
    const int lane = tid & 31, w = tid >> 5;
    __shared__ float red[8];

    const float* eb = e + b * L;
    const float* mb = mask + b * L;

    float vals[4];
    float m = -3.402823e38f;
#pragma unroll
    for (int i = 0; i < 4; ++i) { vals[i] = eb[tid + i * 256]; m = fmaxf(m, vals[i]); }
#pragma unroll
    for (int o = 16; o > 0; o >>= 1) m = fmaxf(m, __shfl_down(m, o, 32));
    if (lane == 0) red[w] = m;
    __syncthreads();
    float bm = red[0];
#pragma unroll
    for (int i = 1; i < 8; ++i) bm = fmaxf(bm, red[i]);
    __syncthreads();

    float p[4], s = 0.f;
#pragma unroll
    for (int i = 0; i < 4; ++i) {
        p[i] = __expf(vals[i] - bm) * mb[tid + i * 256];
        s += p[i];
    }
#pragma unroll
    for (int o = 16; o > 0; o >>= 1) s += __shfl_down(s, o, 32);
    if (lane == 0) red[w] = s;
    __syncthreads();
    float ts = 0.f;
#pragma unroll
    for (int i = 0; i < 8; ++i) ts += red[i];
    const float inv = 1.0f / ts;
#pragma unroll
    for (int i = 0; i < 4; ++i) attn[b * L + tid + i * 256] = p[i] * inv;
}

// ---------------------------------------------------------------------------
// Phase 4a: partial c_t over an L-chunk; grid (B, LCHUNKS) for HBM saturation.
// Each thread owns 4 consecutive h (float4), loops 64 l's. Deterministic
// (no float atomics). Streams 256 MB.
// ---------------------------------------------------------------------------
__global__ __launch_bounds__(256)
void weighted_sum_partial(const float* __restrict__ attn, const float* __restrict__ enc,
                          float* __restrict__ part) {
    const int b = blockIdx.x;
    const int chunk = blockIdx.y;
    const int tid = threadIdx.x;          // h4 index, 0..255
    const int l0 = chunk * LSUB;

    const float* ab = attn + b * L + l0;
    const float4* eb = (const float4*)(enc + ((long long)b * L + l0) * H2);

    float4 acc = make_float4(0.f, 0.f, 0.f, 0.f);
    for (int l = 0; l < LSUB; ++l) {
        const float a = ab[l];
        float4 x = eb[(long long)l * (H2 / 4) + tid];
        acc.x += a * x.x; acc.y += a * x.y; acc.z += a * x.z; acc.w += a * x.w;
    }
    float4* pb = (float4*)(part + ((long long)chunk * B + b) * H2);
    pb[tid] = acc;
}

// Phase 4b: reduce the 16 partials into d_out.
__global__ __launch_bounds__(256)
void reduce_partials(const float* __restrict__ part, float* __restrict__ out) {
    const int idx = blockIdx.x * 256 + threadIdx.x;  // over B*H2
    float s = 0.f;
#pragma unroll
    for (int c = 0; c < LCHUNKS; ++c) s += part[c * (B * H2) + idx];
    out[idx] = s;
}

// ---------------------------------------------------------------------------
extern "C" void kernel_launch(void* const* d_in, const int* in_sizes, int n_in,
                              void* d_out, int out_size, void* d_ws, size_t ws_size,
                              hipStream_t stream) {
    const float* s_t_hat  = (const float*)d_in[0];   // [B, H2]
    const float* enc_out  = (const float*)d_in[1];   // [B, L, H2]
    const float* enc_feat = (const float*)d_in[2];   // [B*L, H2]
    const float* mask     = (const float*)d_in[3];   // [B, L]
    const float* W        = (const float*)d_in[4];   // [H2, H2]
    const float* bias     = (const float*)d_in[5];   // [H2]
    const float* v        = (const float*)d_in[6];   // [H2]
    float* out = (float*)d_out;

    float* dec  = (float*)d_ws;                      // B*H2
    float* e    = dec + (size_t)B * H2;              // B*L
    float* attn = e + (size_t)B * L;                 // B*L
    float* part = attn + (size_t)B * L;              // LCHUNKS*B*H2

    dec_gemm_wmma<<<dim3(H2 / 16, B / 16), 32, 0, stream>>>(s_t_hat, W, bias, dec);
    score_kernel<<<(B * L) / 8, 256, 0, stream>>>(enc_feat, dec, v, e);
    softmax_kernel<<<B, 256, 0, stream>>>(e, mask, attn);
    weighted_sum_partial<<<dim3(B, LCHUNKS), 256, 0, stream>>>(attn, enc_out, part);
    reduce_partials<<<(B * H2) / 256, 256, 0, stream>>>(part, out);
}